// RoiPoolingConv_40295383171365
// MI455X (gfx1250) — compile-verified
//
#include <hip/hip_runtime.h>
#include <cstdint>

namespace {
constexpr int kPool = 7;
constexpr int kH = 128;
constexpr int kW = 128;
constexpr int kC = 1024;
constexpr float kScale = 1.0f / 16.0f;
}  // namespace

// One block per (roi, bin). 256 threads, 4 channels (one float4) per thread.
// The 4 corner channel-vectors are staged into LDS with CDNA5 async
// global->LDS B128 loads (ASYNCcnt), then the bilinear lerp is applied and the
// result stored with one coalesced B128 per thread.
__global__ __launch_bounds__(256)
void roi_align_bilinear_async(const float* __restrict__ img,
                              const float* __restrict__ rois,
                              float* __restrict__ out) {
  __shared__ float smem[4 * kC];  // 16 KB: 4 corner pixels x 1024 channels

  const int bin = blockIdx.x;   // 0..48
  const int roi = blockIdx.y;   // 0..255
  const int iy  = bin / kPool;
  const int ix  = bin - iy * kPool;
  const int t   = threadIdx.x;  // 0..255
  const int c   = t << 2;       // channel base for this thread

  // ROI box -> feature-map coords (matches reference op-for-op, f32).
  const float4 r = reinterpret_cast<const float4*>(rois)[roi];
  const int x0 = (int)(r.x * kScale);
  const int y0 = (int)(r.y * kScale);
  const int w  = (int)(r.z * kScale);
  const int h  = (int)(r.w * kScale);

  const float sy = (float)iy * ((float)h / (float)kPool);
  const float sx = (float)ix * ((float)w / (float)kPool);
  const float fy = floorf(sy);
  const float fx = floorf(sx);
  const float ty = sy - fy;
  const float tx = sx - fx;
  const int y_lo = (int)fy;
  const int x_lo = (int)fx;
  const int y_hi = min(y_lo + 1, max(h - 1, 0));
  const int x_hi = min(x_lo + 1, max(w - 1, 0));
  const int gy0 = min(max(y0 + y_lo, 0), kH - 1);
  const int gy1 = min(max(y0 + y_hi, 0), kH - 1);
  const int gx0 = min(max(x0 + x_lo, 0), kW - 1);
  const int gx1 = min(max(x0 + x_hi, 0), kW - 1);

  // Global byte addresses of this thread's float4 inside each corner pixel.
  const uint64_t base =
      (uint64_t)(uintptr_t)img + (uint64_t)(unsigned)c * 4ull;
  const uint64_t a00 = base + (uint64_t)((gy0 * kW + gx0) * kC) * 4ull;
  const uint64_t a01 = base + (uint64_t)((gy0 * kW + gx1) * kC) * 4ull;
  const uint64_t a10 = base + (uint64_t)((gy1 * kW + gx0) * kC) * 4ull;
  const uint64_t a11 = base + (uint64_t)((gy1 * kW + gx1) * kC) * 4ull;

  // LDS byte offsets (low 32 bits of the generic pointer == LDS offset).
  const unsigned l00 = (unsigned)(uintptr_t)(&smem[0 * kC + c]);
  const unsigned l01 = (unsigned)(uintptr_t)(&smem[1 * kC + c]);
  const unsigned l10 = (unsigned)(uintptr_t)(&smem[2 * kC + c]);
  const unsigned l11 = (unsigned)(uintptr_t)(&smem[3 * kC + c]);

  // CDNA5 async global->LDS gather (per-lane 16B), tracked on ASYNCcnt.
  asm volatile("global_load_async_to_lds_b128 %0, %1, off"
               :: "v"(l00), "v"(a00) : "memory");
  asm volatile("global_load_async_to_lds_b128 %0, %1, off"
               :: "v"(l01), "v"(a01) : "memory");
  asm volatile("global_load_async_to_lds_b128 %0, %1, off"
               :: "v"(l10), "v"(a10) : "memory");
  asm volatile("global_load_async_to_lds_b128 %0, %1, off"
               :: "v"(l11), "v"(a11) : "memory");
  // Each thread consumes only the data it staged itself -> a per-wave
  // asynccnt wait is sufficient; no workgroup barrier needed.
  asm volatile("s_wait_asynccnt 0" ::: "memory");

  const float4 v00 = *reinterpret_cast<const float4*>(&smem[0 * kC + c]);
  const float4 v01 = *reinterpret_cast<const float4*>(&smem[1 * kC + c]);
  const float4 v10 = *reinterpret_cast<const float4*>(&smem[2 * kC + c]);
  const float4 v11 = *reinterpret_cast<const float4*>(&smem[3 * kC + c]);

  float4 o;
  {
    float top, bot;
    top = v00.x + tx * (v01.x - v00.x);
    bot = v10.x + tx * (v11.x - v10.x);
    o.x = top + ty * (bot - top);
    top = v00.y + tx * (v01.y - v00.y);
    bot = v10.y + tx * (v11.y - v10.y);
    o.y = top + ty * (bot - top);
    top = v00.z + tx * (v01.z - v00.z);
    bot = v10.z + tx * (v11.z - v10.z);
    o.z = top + ty * (bot - top);
    top = v00.w + tx * (v01.w - v00.w);
    bot = v10.w + tx * (v11.w - v10.w);
    o.w = top + ty * (bot - top);
  }

  const size_t obase =
      ((size_t)roi * (kPool * kPool) + (size_t)bin) * kC + (size_t)c;
  *reinterpret_cast<float4*>(out + obase) = o;
}

extern "C" void kernel_launch(void* const* d_in, const int* in_sizes, int n_in,
                              void* d_out, int out_size, void* d_ws,
                              size_t ws_size, hipStream_t stream) {
  (void)in_sizes; (void)n_in; (void)out_size; (void)d_ws; (void)ws_size;
  const float* img  = (const float*)d_in[0];   // (1,128,128,1024) f32
  const float* rois = (const float*)d_in[1];   // (1,256,4) f32
  float* out = (float*)d_out;                  // (1,256,7,7,1024) f32

  dim3 grid(kPool * kPool, 256);  // (49 bins, 256 rois)
  dim3 block(256);
  roi_align_bilinear_async<<<grid, block, 0, stream>>>(img, rois, out);
}